// HybridSampler_67997922230582
// MI455X (gfx1250) — compile-verified
//
#include <hip/hip_runtime.h>
#include <stdint.h>
#include <math.h>

// ---------------- problem constants (from reference) ----------------
#define DX 100
#define DY 100
#define DZ 8
#define N_VOX (DX * DY * DZ)       // 80000
#define SEGS  (2 * N_VOX)          // 160000
#define PTS_TILE 32                // async-path tile (points)
#define TILE_PTS 128               // TDM-path tile (points) = 2KB
#define WAVES_PER_BLOCK 8

typedef unsigned int       u32;
typedef unsigned long long u64;
typedef u32   u32v4 __attribute__((ext_vector_type(4)));
typedef u32   u32v8 __attribute__((ext_vector_type(8)));
typedef float v2f   __attribute__((ext_vector_type(2)));
typedef float v8f   __attribute__((ext_vector_type(8)));

#if __has_builtin(__builtin_amdgcn_wmma_f32_16x16x4_f32)
#define HAVE_WMMA_F32_4 1
#pragma message("PROBE: has __builtin_amdgcn_wmma_f32_16x16x4_f32")
#else
#define HAVE_WMMA_F32_4 0
#pragma message("PROBE: MISSING __builtin_amdgcn_wmma_f32_16x16x4_f32")
#endif

// ---------------- async-to-LDS helpers ----------------
__device__ __forceinline__ void async_ld16(const float4* g, const float4* lds_dst) {
  unsigned int lds = (unsigned int)(uintptr_t)lds_dst;   // low 32b of generic = LDS offset
  unsigned long long ga = (unsigned long long)(uintptr_t)g;
  asm volatile("global_load_async_to_lds_b128 %0, %1, off"
               :: "v"(lds), "v"(ga) : "memory");
}
__device__ __forceinline__ void wait_async_le1() { asm volatile("s_wait_asynccnt 1" ::: "memory"); }
__device__ __forceinline__ void wait_async_le0() { asm volatile("s_wait_asynccnt 0" ::: "memory"); }

// ---------------- TDM (Tensor Data Mover) helpers ----------------
#define TDM_TILE_FLOATS (TILE_PTS * 4)   // 512 f32 elements = 2KB per tile

__device__ __forceinline__ void tdm_load_tile(u64 gaddr, u32 lds_addr) {
  u32v4 g0;
  g0[0] = 1u;                                    // count=1, is_restore=0, gather=0
  g0[1] = lds_addr;                              // LDS byte address
  g0[2] = (u32)gaddr;                            // global_addr[31:0]  (tile start)
  g0[3] = (u32)(gaddr >> 32) | (2u << 30);       // global_addr[56:32] | type=2 ("image")
  u32v8 g1;
  g1[0] = 2u << 16;                              // workgroup_mask=0 | data_size=2 (4B)
  g1[1] = (TDM_TILE_FLOATS & 0xffffu) << 16;     // tensor_dim0[15:0]
  g1[2] = (TDM_TILE_FLOATS >> 16) | (1u << 16);  // tensor_dim0[31:16] | tensor_dim1=1
  g1[3] = (TDM_TILE_FLOATS & 0xffffu) << 16;     // tile_dim0=512
  g1[4] = 0u;                                    // tile_dim1=0, tile_dim2=0
  g1[5] = TDM_TILE_FLOATS;                       // tensor_dim0_stride[31:0]
  g1[6] = 0u;
  g1[7] = 0u;
  asm volatile("tensor_load_to_lds %0, %1" :: "s"(g0), "s"(g1) : "memory");
}
__device__ __forceinline__ void wait_tensor_le1() { asm volatile("s_wait_tensorcnt 0x1" ::: "memory"); }
__device__ __forceinline__ void wait_tensor_le0() { asm volatile("s_wait_tensorcnt 0x0" ::: "memory"); }

// ---------------- tiny float3 helpers ----------------
struct F3 { float x, y, z; };
__device__ __forceinline__ F3 f3(float a, float b, float c) { F3 r; r.x=a; r.y=b; r.z=c; return r; }
__device__ __forceinline__ F3 cross3(F3 a, F3 b) {
  return f3(a.y*b.z - a.z*b.y, a.z*b.x - a.x*b.z, a.x*b.y - a.y*b.x);
}
__device__ __forceinline__ float dot3(F3 a, F3 b) { return a.x*b.x + a.y*b.y + a.z*b.z; }

// ---------------- phase 1: moment scatter (f32 atomics, L2-resident bins) ----------------
__device__ __forceinline__ void accum_point(float4 p, float* __restrict__ out) {
  int b  = (int)p.x;
  int cx = (int)floorf(p.y + 50.0f);
  int cy = (int)floorf(p.z + 50.0f);
  int cz = (int)floorf(p.w + 5.0f);
  cx = min(max(cx, 0), DX - 1);
  cy = min(max(cy, 0), DY - 1);
  cz = min(max(cz, 0), DZ - 1);
  int pid = b * N_VOX + (cx * DY + cy) * DZ + cz;
  float* r = out + (size_t)pid * 16;
  atomicAdd(r + 0, 1.0f);
  atomicAdd(r + 1, p.y);
  atomicAdd(r + 2, p.z);
  atomicAdd(r + 3, p.w);
  atomicAdd(r + 4, p.y * p.y);
  atomicAdd(r + 5, p.y * p.z);
  atomicAdd(r + 6, p.y * p.w);
  atomicAdd(r + 7, p.z * p.z);
  atomicAdd(r + 8, p.z * p.w);
  atomicAdd(r + 9, p.w * p.w);
}

// Primary: TDM-streamed tiles (TENSORcnt), double-buffered per wave.
__global__ void __launch_bounds__(256)
k_accum_tdm(const float4* __restrict__ pts, int np, float* __restrict__ out) {
  __shared__ float4 stage[WAVES_PER_BLOCK][2][TILE_PTS];   // 32 KB
  const int lane  = threadIdx.x & 31;
  const int wave  = threadIdx.x >> 5;
  const int swave = __builtin_amdgcn_readfirstlane(wave);
  const int gwave = blockIdx.x * WAVES_PER_BLOCK + swave;
  const int nwave = gridDim.x * WAVES_PER_BLOCK;
  const int ntiles = np / TILE_PTS;

  const u64 base = (u64)(uintptr_t)pts;
  const u32 lds0 =
      (u32)__builtin_amdgcn_readfirstlane((int)(u32)(uintptr_t)&stage[0][0][0])
      + (u32)swave * (2u * TILE_PTS * (u32)sizeof(float4));

  if (gwave < ntiles)
    tdm_load_tile(base + (u64)gwave * (TILE_PTS * 16u), lds0);

  int buf = 0;
  for (int t = gwave; t < ntiles; t += nwave) {
    const int tn = t + nwave;
    if (tn < ntiles) {
      tdm_load_tile(base + (u64)tn * (TILE_PTS * 16u),
                    lds0 + (u32)(buf ^ 1) * (TILE_PTS * 16u));
      wait_tensor_le1();
    } else {
      wait_tensor_le0();
    }
    const float4* sb = stage[wave][buf];
#pragma unroll
    for (int j = 0; j < TILE_PTS / 32; ++j)
      accum_point(sb[j * 32 + lane], out);
    buf ^= 1;
  }
  for (int i = ntiles * TILE_PTS + gwave * 32 + lane; i < np; i += nwave * 32)
    accum_point(pts[i], out);
}

// Backup implementation: ASYNCcnt b128 pipeline.
__global__ void __launch_bounds__(256)
k_accum_async(const float4* __restrict__ pts, int np, float* __restrict__ out) {
  __shared__ float4 stage[WAVES_PER_BLOCK][2][PTS_TILE];
  const int lane  = threadIdx.x & 31;
  const int wave  = threadIdx.x >> 5;
  const int gwave = blockIdx.x * WAVES_PER_BLOCK + wave;
  const int nwave = gridDim.x * WAVES_PER_BLOCK;
  const int ntiles = np / PTS_TILE;

  if (gwave < ntiles)
    async_ld16(&pts[(size_t)gwave * PTS_TILE + lane], &stage[wave][0][lane]);
  int buf = 0;
  for (int t = gwave; t < ntiles; t += nwave) {
    const int tn = t + nwave;
    if (tn < ntiles) {
      async_ld16(&pts[(size_t)tn * PTS_TILE + lane], &stage[wave][buf ^ 1][lane]);
      wait_async_le1();
    } else {
      wait_async_le0();
    }
    float4 p = stage[wave][buf][lane];
    accum_point(p, out);
    buf ^= 1;
  }
  for (int i = ntiles * PTS_TILE + gwave * 32 + lane; i < np; i += nwave * 32)
    accum_point(pts[i], out);
}

// ---------------- analytic symmetric 3x3 eigensolver ----------------
__device__ __forceinline__ F3 eigvec3(float c00, float c01, float c02,
                                      float c11, float c12, float c22,
                                      float lam, float degen_thresh, bool& ok) {
  F3 r0 = f3(c00 - lam, c01, c02);
  F3 r1 = f3(c01, c11 - lam, c12);
  F3 r2 = f3(c02, c12, c22 - lam);
  F3 a = cross3(r0, r1);
  F3 b = cross3(r0, r2);
  F3 c = cross3(r1, r2);
  float na = dot3(a, a), nb = dot3(b, b), nc = dot3(c, c);
  F3 best = a; float nbest = na;
  if (nb > nbest) { best = b; nbest = nb; }
  if (nc > nbest) { best = c; nbest = nc; }
  ok = nbest > degen_thresh;
  float s = rsqrtf(fmaxf(nbest, 1e-38f));
  return f3(best.x * s, best.y * s, best.z * s);
}

__device__ __forceinline__ void eigh3(float c00, float c01, float c02,
                                      float c11, float c12, float c22,
                                      float& l0, float& l1, float& l2,
                                      F3& v0, F3& v1, F3& v2) {
  const float p1 = c01 * c01 + c02 * c02 + c12 * c12;
  const float q  = (c00 + c11 + c22) * (1.0f / 3.0f);
  const float d0 = c00 - q, d1 = c11 - q, d2 = c22 - q;
  const float p2 = d0 * d0 + d1 * d1 + d2 * d2 + 2.0f * p1;
  const float frob2 = c00 * c00 + c11 * c11 + c22 * c22 + 2.0f * p1;

  if (p2 <= 1e-18f * frob2 + 1e-30f) {
    l0 = l1 = l2 = q;
    v0 = f3(1, 0, 0); v1 = f3(0, 1, 0); v2 = f3(0, 0, 1);
    return;
  }
  const float pp = sqrtf(p2 * (1.0f / 6.0f));
  const float ip = 1.0f / pp;
  const float b00 = d0 * ip, b11 = d1 * ip, b22 = d2 * ip;
  const float b01 = c01 * ip, b02 = c02 * ip, b12 = c12 * ip;
  float detB = b00 * (b11 * b22 - b12 * b12)
             - b01 * (b01 * b22 - b12 * b02)
             + b02 * (b01 * b12 - b11 * b02);
  float r = fminf(1.0f, fmaxf(-1.0f, 0.5f * detB));
  const float phi = acosf(r) * (1.0f / 3.0f);
  l2 = q + 2.0f * pp * cosf(phi);
  l0 = q + 2.0f * pp * cosf(phi + 2.0943951023931953f);
  l1 = 3.0f * q - l0 - l2;

  const float dth = 1e-16f * frob2 * frob2 + 1e-32f;
  bool ok2, ok0;
  v2 = eigvec3(c00, c01, c02, c11, c12, c22, l2, dth, ok2);
  if (!ok2) v2 = f3(0, 0, 1);
  v0 = eigvec3(c00, c01, c02, c11, c12, c22, l0, dth, ok0);
  float d = dot3(v0, v2);
  v0 = f3(v0.x - d * v2.x, v0.y - d * v2.y, v0.z - d * v2.z);
  float n0 = dot3(v0, v0);
  if (!ok0 || n0 < 1e-12f) {
    F3 u = (fabsf(v2.x) < 0.9f) ? f3(1, 0, 0) : f3(0, 1, 0);
    float du = dot3(u, v2);
    v0 = f3(u.x - du * v2.x, u.y - du * v2.y, u.z - du * v2.z);
    n0 = dot3(v0, v0);
  }
  float is0 = rsqrtf(n0);
  v0 = f3(v0.x * is0, v0.y * is0, v0.z * is0);
  v1 = cross3(v2, v0);
}

// ssq slot for cov entry (r,c): 4:xx 5:xy 6:xz 7:yy 8:yz 9:zz
__device__ __forceinline__ int symidx(int r, int c) {
  int m = min(r, c), M = max(r, c);
  return (m == 0) ? (4 + M) : ((m == 1) ? (6 + M) : 9);
}

// ---------------- phase 2 (primary): WMMA rank-1 covariance finalize ----------------
// One wave = 4 segments. u = v = [mean(s0)|mean(s1)|mean(s2)|mean(s3)] (4 comps each,
// 4th = 0); V_WMMA_F32_16X16X4_F32 with only K=0 live computes D = u*v^T, whose
// diagonal 4x4 blocks are the per-segment mean outer products. D layout: lane N holds
// column N (rows in d[0..7]); block a columns sit at lanes {4a+c} (a<2) / {16+4a+c}
// (a>=2) with rows in d[(a&1)*4 + r].
__global__ void __launch_bounds__(256)
k_finalize_wmma(float* __restrict__ out) {
  const int lane = threadIdx.x & 31;
  const int gw   = (int)((blockIdx.x * blockDim.x + threadIdx.x) >> 5);
  const int seg_base = gw * 4;

  // ---- build rank-1 operands (A-matrix VGPR0/lanes0-15 = K=0 column) ----
  int iseg = min(seg_base + (lane >> 2), SEGS - 1);
  const float* irow = out + (size_t)iseg * 16;
  float iinv  = 1.0f / fmaxf(irow[0], 1.0f);
  int   icomp = lane & 3;
  float mu = (lane < 16 && icomp < 3) ? irow[1 + icomp] * iinv : 0.0f;

  v2f aop; aop.x = (lane < 16) ? mu : 0.0f; aop.y = 0.0f;  // K=1..3 slots = 0
  v2f bop = aop;                                           // symmetric rank-1
  v8f cop = {};

  // ---- per-(segment, column) assignment matching the D layout ----
  const bool active = (lane < 8) || (lane >= 24);
  const int  a    = ((lane < 8) ? (lane >> 2) : ((lane >> 2) - 4)) & 3;
  const int  c    = lane & 3;
  const int  ceff = min(c, 2);
  const int  seg  = min(seg_base + a, SEGS - 1);
  const float* row = out + (size_t)seg * 16;
  const float cnt = row[0];
  const float inv = 1.0f / fmaxf(cnt, 1.0f);
  const float m_c = row[1 + ceff] * inv;                   // mean component `ceff`

#if HAVE_WMMA_F32_4
  v8f d = __builtin_amdgcn_wmma_f32_16x16x4_f32(
      /*neg_a=*/false, aop, /*neg_b=*/false, bop,
      /*c_mod=*/(short)0, cop, /*reuse_a=*/false, /*reuse_b=*/false);
  float o0 = d[(a & 1) * 4 + 0];
  float o1 = d[(a & 1) * 4 + 1];
  float o2 = d[(a & 1) * 4 + 2];
#else
  (void)aop; (void)bop; (void)cop;
  float mr0 = __shfl(m_c, (lane & ~3) + 0);
  float mr1 = __shfl(m_c, (lane & ~3) + 1);
  float mr2 = __shfl(m_c, (lane & ~3) + 2);
  float o0 = mr0 * m_c, o1 = mr1 * m_c, o2 = mr2 * m_c;
#endif

  // cov column `ceff` = ssq_col/denom - outer_col (+ jitter on diagonal)
  float q0 = row[symidx(0, ceff)] * inv - o0;
  float q1 = row[symidx(1, ceff)] * inv - o1;
  float q2 = row[symidx(2, ceff)] * inv - o2;
  if      (ceff == 0) q0 += 1.0e-4f;
  else if (ceff == 1) q1 += 2.0e-4f;
  else                q2 += 3.0e-4f;

  // gather symmetric 3x3 + mean to the column-0 lane of each segment
  float n_q1  = __shfl(q1,  lane + 1);
  float n_q2a = __shfl(q2,  lane + 1);
  float n_q2b = __shfl(q2,  lane + 2);
  float n_m1  = __shfl(m_c, lane + 1);
  float n_m2  = __shfl(m_c, lane + 2);

  if (active && c == 0 && (seg_base + a) < SEGS) {
    float l0, l1, l2; F3 e0, e1, e2;
    eigh3(q0, q1, q2, n_q1, n_q2a, n_q2b, l0, l1, l2, e0, e1, e2);
    float4* o4 = (float4*)(out + (size_t)(seg_base + a) * 16);
    o4[0] = make_float4(cnt, m_c, n_m1, n_m2);
    o4[1] = make_float4(l0, l1, l2, e0.x);
    o4[2] = make_float4(e1.x, e2.x, e0.y, e1.y);
    o4[3] = make_float4(e2.y, e0.z, e1.z, e2.z);
  }
}

// Backup: per-lane finalize (one segment per lane).
__global__ void __launch_bounds__(256)
k_finalize(float* __restrict__ out) {
  const int s = blockIdx.x * blockDim.x + threadIdx.x;
  if (s >= SEGS) return;
  float* row = out + (size_t)s * 16;
  const float cnt = row[0];
  const float inv = 1.0f / fmaxf(cnt, 1.0f);
  const float mx = row[1] * inv, my = row[2] * inv, mz = row[3] * inv;
  float c00 = row[4] * inv - mx * mx + 1.0e-4f;
  float c01 = row[5] * inv - mx * my;
  float c02 = row[6] * inv - mx * mz;
  float c11 = row[7] * inv - my * my + 2.0e-4f;
  float c12 = row[8] * inv - my * mz;
  float c22 = row[9] * inv - mz * mz + 3.0e-4f;
  float l0, l1, l2; F3 e0, e1, e2;
  eigh3(c00, c01, c02, c11, c12, c22, l0, l1, l2, e0, e1, e2);
  float4* o4 = (float4*)row;
  o4[0] = make_float4(cnt, mx, my, mz);
  o4[1] = make_float4(l0, l1, l2, e0.x);
  o4[2] = make_float4(e1.x, e2.x, e0.y, e1.y);
  o4[3] = make_float4(e2.y, e0.z, e1.z, e2.z);
}

// ---------------- host launcher ----------------
extern "C" void kernel_launch(void* const* d_in, const int* in_sizes, int n_in,
                              void* d_out, int out_size, void* d_ws, size_t ws_size,
                              hipStream_t stream) {
  (void)n_in; (void)out_size; (void)d_ws; (void)ws_size;
  const float4* pts = (const float4*)d_in[0];   // bxyz: [N,4] float32
  const int np = in_sizes[0] / 4;
  float* out = (float*)d_out;                   // [160000, 16] float32

  hipMemsetAsync(out, 0, (size_t)SEGS * 16 * sizeof(float), stream);
  k_accum_tdm<<<2048, 256, 0, stream>>>(pts, np, out);
  // 4 segments per wave, 8 waves per block -> 32 segments per block
  k_finalize_wmma<<<(SEGS + 31) / 32, 256, 0, stream>>>(out);
}